// GAT_62955630624875
// MI455X (gfx1250) — compile-verified
//
#include <hip/hip_runtime.h>
#include <hip/hip_bf16.h>

#define NNODES 50000
#define NEDGES 800000
#define ETOT   (NEDGES + NNODES)
#define IN_CH  128
#define H1H    4
#define C1C    64
#define HC1    (H1H * C1C)   /* 256 */
#define OUT_CH 128
#define NEG_SLOPE 0.2f

typedef float v2f __attribute__((ext_vector_type(2)));
typedef float v8f __attribute__((ext_vector_type(8)));

// ---------------------------------------------------------------------------
// Monotonic float <-> uint mapping for atomicMax on floats
// ---------------------------------------------------------------------------
__device__ __forceinline__ unsigned encf(float f) {
    unsigned u = __float_as_uint(f);
    return (u >> 31) ? ~u : (u | 0x80000000u);
}
__device__ __forceinline__ float decf(unsigned e) {
    return __uint_as_float((e >> 31) ? (e ^ 0x80000000u) : ~e);
}

// ---------------------------------------------------------------------------
// FP32 WMMA GEMM: C[M,NC] = A[M,K] @ B[K,NC]
// Block = 256 threads = 8 waves; block covers 16 rows x 128 cols of C.
// The 16 x K A tile is staged in LDS once (coalesced) and shared by all 8
// waves; each wave computes one 16x16 tile with V_WMMA_F32_16X16X4_F32.
// LDS row stride K+2 floats: keeps 8B alignment for ds_load_b64 and spreads
// the 16 reading lanes across distinct even banks (l*130 mod 64 = 2l).
// A-frag (16x4 f32): lanes 0-15 hold K={0,1}, lanes 16-31 hold K={2,3}, M=l.
// C/D layout: VGPR i -> M=i (lanes 0-15) / M=8+i (lanes 16-31), N = lane&15.
// ---------------------------------------------------------------------------
template <int K, int NC>
__global__ __launch_bounds__(256)
void gemm_f32_wmma(const float* __restrict__ A, const float* __restrict__ B,
                   float* __restrict__ C) {
    constexpr int LDA = K + 2;
    __shared__ float As[16 * LDA];

    const int tid  = threadIdx.x;
    const int lane = tid & 31;
    const int wave = tid >> 5;
    constexpr int blkPerRow = NC / 128;            // col-tile groups per row
    const int tM = blockIdx.x / blkPerRow;
    const int tN = (blockIdx.x % blkPerRow) * 8 + wave;

    // cooperative, coalesced load of the 16 x K A tile into LDS
    const float* arow = A + (size_t)tM * 16 * K;
    #pragma unroll
    for (int i = tid; i < 16 * K; i += 256) {
        int r  = i / K;            // K is a power of two -> shifts
        int kk = i - r * K;
        As[r * LDA + kk] = arow[i];
    }
    __syncthreads();

    const int half = lane >> 4;    // 0: K pair {0,1}, 1: K pair {2,3}
    const int l    = lane & 15;
    const float* ap = &As[l * LDA + 2 * half];
    const float* bp = B + (size_t)(2 * half) * NC + tN * 16 + l;

    v8f acc = {0.f, 0.f, 0.f, 0.f, 0.f, 0.f, 0.f, 0.f};

    #pragma unroll 8
    for (int k0 = 0; k0 < K; k0 += 4) {
        v2f a = *(const v2f*)(ap + k0);            // ds_load_b64
        v2f b;
        b.x = bp[(size_t)k0 * NC];
        b.y = bp[(size_t)(k0 + 1) * NC];
        acc = __builtin_amdgcn_wmma_f32_16x16x4_f32(
            false, a, false, b, (short)0, acc, false, false);
    }

    float* cp = C + (size_t)(tM * 16 + half * 8) * NC + tN * 16 + l;
    #pragma unroll
    for (int i = 0; i < 8; ++i) cp[(size_t)i * NC] = acc[i];
}

// ---------------------------------------------------------------------------
// Attention dot products: as[n,h] = sum_c h[n,h,c]*att_s[h,c]  (same for ad)
// ---------------------------------------------------------------------------
template <int H, int C>
__global__ __launch_bounds__(256)
void attn_dots(const float* __restrict__ h, const float* __restrict__ atts,
               const float* __restrict__ attd, float* __restrict__ as_,
               float* __restrict__ ad_, int n) {
    int t = blockIdx.x * blockDim.x + threadIdx.x;
    if (t >= n * H) return;
    int node = t / H, hh = t - node * H;
    const float* hp = h + (size_t)node * (H * C) + hh * C;
    const float* sp = atts + hh * C;
    const float* dp = attd + hh * C;
    float s = 0.f, d = 0.f;
    #pragma unroll 8
    for (int c = 0; c < C; ++c) {
        float v = hp[c];
        s += v * sp[c];
        d += v * dp[c];
    }
    as_[t] = s;
    ad_[t] = d;
}

// ---------------------------------------------------------------------------
// Pass 1: e = leaky_relu(a_s[src]+a_d[dst]); store e; segment max via atomicMax
// ---------------------------------------------------------------------------
template <int H>
__global__ __launch_bounds__(256)
void edge_logits(const int* __restrict__ srcI, const int* __restrict__ dstI,
                 const float* __restrict__ as_, const float* __restrict__ ad_,
                 float* __restrict__ ebuf, unsigned* __restrict__ emax) {
    int t = blockIdx.x * blockDim.x + threadIdx.x;
    if (t >= ETOT * H) return;
    int k = t / H, hh = t - k * H;
    int s = (k < NEDGES) ? srcI[k] : (k - NEDGES);
    int d = (k < NEDGES) ? dstI[k] : (k - NEDGES);
    float e = as_[s * H + hh] + ad_[d * H + hh];
    e = (e > 0.f) ? e : NEG_SLOPE * e;
    ebuf[t] = e;
    atomicMax(&emax[d * H + hh], encf(e));
}

// ---------------------------------------------------------------------------
// Pass 2: p = exp(e - max[dst]); store p; segment sum via atomicAdd
// ---------------------------------------------------------------------------
template <int H>
__global__ __launch_bounds__(256)
void edge_exp(const int* __restrict__ srcI, const int* __restrict__ dstI,
              float* __restrict__ ebuf, const unsigned* __restrict__ emax,
              float* __restrict__ denom) {
    int t = blockIdx.x * blockDim.x + threadIdx.x;
    if (t >= ETOT * H) return;
    int k = t / H, hh = t - k * H;
    int d = (k < NEDGES) ? dstI[k] : (k - NEDGES);
    float m = decf(emax[d * H + hh]);
    float p = expf(ebuf[t] - m);
    ebuf[t] = p;
    atomicAdd(&denom[d * H + hh], p);
}

// ---------------------------------------------------------------------------
// Pass 3: out[dst, j..j+3] += h[src, j..j+3] * alpha.  One lane handles 4
// consecutive features of one edge: a wave reads a contiguous 512B slice of
// h[src] (global_load_b128) and issues 4 coalesced atomicAdds into agg[dst].
// Everything is L2-resident (h and agg together < 192MB L2).
// ---------------------------------------------------------------------------
template <int H, int HC>
__global__ __launch_bounds__(256)
void edge_message(const int* __restrict__ srcI, const int* __restrict__ dstI,
                  const float* __restrict__ ebuf, const float* __restrict__ denom,
                  const float* __restrict__ hfeat, float* __restrict__ agg) {
    constexpr int Q = HC / 4;                      // float4 groups per edge
    long long t = (long long)blockIdx.x * blockDim.x + threadIdx.x;
    if (t >= (long long)ETOT * Q) return;
    int k = (int)(t / Q);                          // Q power of two -> shift
    int q = (int)(t - (long long)k * Q);
    int j = q * 4;
    int s = (k < NEDGES) ? srcI[k] : (k - NEDGES);
    int d = (k < NEDGES) ? dstI[k] : (k - NEDGES);
    constexpr int C = HC / H;
    int hh = j / C;
    float w = ebuf[k * H + hh] / (denom[d * H + hh] + 1e-16f);
    const float4 v = *(const float4*)&hfeat[(size_t)s * HC + j];
    float* ap = &agg[(size_t)d * HC + j];
    atomicAdd(ap + 0, v.x * w);
    atomicAdd(ap + 1, v.y * w);
    atomicAdd(ap + 2, v.z * w);
    atomicAdd(ap + 3, v.w * w);
}

// ---------------------------------------------------------------------------
// x = relu(x + bias) in place (COLS is a power of two)
// ---------------------------------------------------------------------------
template <int COLS>
__global__ __launch_bounds__(256)
void bias_relu(float* __restrict__ x, const float* __restrict__ b, long long n) {
    long long t = (long long)blockIdx.x * blockDim.x + threadIdx.x;
    if (t >= n) return;
    float v = x[t] + b[(int)(t & (COLS - 1))];
    x[t] = (v > 0.f) ? v : 0.f;
}

// ---------------------------------------------------------------------------
extern "C" void kernel_launch(void* const* d_in, const int* in_sizes, int n_in,
                              void* d_out, int out_size, void* d_ws, size_t ws_size,
                              hipStream_t stream) {
    const float* x     = (const float*)d_in[0];
    const int*   ei    = (const int*)d_in[1];
    const int*   srcI  = ei;                // edge_index[0]
    const int*   dstI  = ei + NEDGES;       // edge_index[1]
    const float* W1    = (const float*)d_in[2];
    const float* atts1 = (const float*)d_in[3];
    const float* attd1 = (const float*)d_in[4];
    const float* b1    = (const float*)d_in[5];
    const float* W2    = (const float*)d_in[6];
    const float* atts2 = (const float*)d_in[7];
    const float* attd2 = (const float*)d_in[8];
    const float* b2    = (const float*)d_in[9];
    float* out = (float*)d_out;

    // Workspace carve-up
    char* ws = (char*)d_ws;
    size_t off = 0;
    auto take = [&](size_t bytes) -> char* {
        char* p = ws + off;
        off += (bytes + 255) & ~(size_t)255;
        return p;
    };
    float*    h1     = (float*)take((size_t)NNODES * HC1 * 4);   // also reused as h2
    float*    agg1   = (float*)take((size_t)NNODES * HC1 * 4);
    float*    as1    = (float*)take((size_t)NNODES * H1H * 4);
    float*    ad1    = (float*)take((size_t)NNODES * H1H * 4);
    unsigned* emax1  = (unsigned*)take((size_t)NNODES * H1H * 4);
    float*    denom1 = (float*)take((size_t)NNODES * H1H * 4);
    float*    ebuf1  = (float*)take((size_t)ETOT * H1H * 4);
    float*    as2    = (float*)take((size_t)NNODES * 4);
    float*    ad2    = (float*)take((size_t)NNODES * 4);
    unsigned* emax2  = (unsigned*)take((size_t)NNODES * 4);
    float*    denom2 = (float*)take((size_t)NNODES * 4);
    float*    ebuf2  = (float*)take((size_t)ETOT * 4);
    float*    h2     = h1;  // layer-1 features dead once layer-1 messages done

    // Zero / init accumulators (enc-space 0 == "-inf" sentinel for emax:
    // every (node, head) is touched at least once thanks to self-loops)
    hipMemsetAsync(agg1,   0, (size_t)NNODES * HC1 * 4, stream);
    hipMemsetAsync(emax1,  0, (size_t)NNODES * H1H * 4, stream);
    hipMemsetAsync(denom1, 0, (size_t)NNODES * H1H * 4, stream);
    hipMemsetAsync(emax2,  0, (size_t)NNODES * 4, stream);
    hipMemsetAsync(denom2, 0, (size_t)NNODES * 4, stream);
    hipMemsetAsync(out,    0, (size_t)NNODES * OUT_CH * 4, stream);

    const int TB = 256;

    // ---- Layer 1 ----
    // h1 = x @ W1  (M=50000, K=128, NC=256): 3125 row-tiles x 2 col-groups
    gemm_f32_wmma<IN_CH, HC1><<<(NNODES / 16) * (HC1 / 128), TB, 0, stream>>>(x, W1, h1);
    attn_dots<H1H, C1C><<<(NNODES * H1H + TB - 1) / TB, TB, 0, stream>>>(
        h1, atts1, attd1, as1, ad1, NNODES);
    edge_logits<H1H><<<(ETOT * H1H + TB - 1) / TB, TB, 0, stream>>>(
        srcI, dstI, as1, ad1, ebuf1, emax1);
    edge_exp<H1H><<<(ETOT * H1H + TB - 1) / TB, TB, 0, stream>>>(
        srcI, dstI, ebuf1, emax1, denom1);
    edge_message<H1H, HC1>
        <<<(unsigned)(((long long)ETOT * (HC1 / 4) + TB - 1) / TB), TB, 0, stream>>>(
        srcI, dstI, ebuf1, denom1, h1, agg1);
    bias_relu<HC1><<<(unsigned)(((long long)NNODES * HC1 + TB - 1) / TB), TB, 0, stream>>>(
        agg1, b1, (long long)NNODES * HC1);

    // ---- Layer 2 ----
    // h2 = agg1 @ W2  (M=50000, K=256, NC=128)
    gemm_f32_wmma<HC1, OUT_CH><<<(NNODES / 16) * (OUT_CH / 128), TB, 0, stream>>>(agg1, W2, h2);
    attn_dots<1, OUT_CH><<<(NNODES + TB - 1) / TB, TB, 0, stream>>>(
        h2, atts2, attd2, as2, ad2, NNODES);
    edge_logits<1><<<(ETOT + TB - 1) / TB, TB, 0, stream>>>(
        srcI, dstI, as2, ad2, ebuf2, emax2);
    edge_exp<1><<<(ETOT + TB - 1) / TB, TB, 0, stream>>>(
        srcI, dstI, ebuf2, emax2, denom2);
    edge_message<1, OUT_CH>
        <<<(unsigned)(((long long)ETOT * (OUT_CH / 4) + TB - 1) / TB), TB, 0, stream>>>(
        srcI, dstI, ebuf2, denom2, h2, out);
    bias_relu<OUT_CH><<<(unsigned)(((long long)NNODES * OUT_CH + TB - 1) / TB), TB, 0, stream>>>(
        out, b2, (long long)NNODES * OUT_CH);
}